// SimplePredictor_27642409517403
// MI455X (gfx1250) — compile-verified
//
#include <hip/hip_runtime.h>
#include <hip/hip_bf16.h>
#include <math.h>

// Problem constants (match reference)
#define BB 8
#define LL 4096
#define DD 2048
#define D2 4096
#define TT 128
#define NSEG 32
#define ROWS_PER_SEG (LL / NSEG)   // 128
#define LN_EPS 1e-5f

typedef __attribute__((ext_vector_type(2))) float v2f;
typedef __attribute__((ext_vector_type(8))) float v8f;

// ---------------------------------------------------------------------------
// Kernel 1: per-batch mask counts (ctx count + tgt count)
// ---------------------------------------------------------------------------
__global__ void count_masks_kernel(const float* __restrict__ tmask,
                                   const float* __restrict__ cmask,
                                   float* __restrict__ counts /* [2*BB] */) {
    __shared__ float red[256];
    const int b = blockIdx.x;
    const int tid = threadIdx.x;

    float cs = 0.f, ts = 0.f;
    for (int l = tid; l < LL; l += 256) {
        cs += cmask[(size_t)b * LL + l];
        ts += tmask[(size_t)b * LL + l];
    }
    // reduce ctx
    red[tid] = cs; __syncthreads();
    for (int s = 128; s > 0; s >>= 1) { if (tid < s) red[tid] += red[tid + s]; __syncthreads(); }
    if (tid == 0) counts[b] = red[0];
    __syncthreads();
    // reduce tgt
    red[tid] = ts; __syncthreads();
    for (int s = 128; s > 0; s >>= 1) { if (tid < s) red[tid] += red[tid + s]; __syncthreads(); }
    if (tid == 0) counts[BB + b] = red[0];
}

// ---------------------------------------------------------------------------
// Kernel 2: phase-1 pooling reduction.  grid = (DD/1024, NSEG, BB), 256 thr.
// Each thread accumulates a float4 over ROWS_PER_SEG rows (fully coalesced
// 4 KB-per-row block loads); writes partial sums to ws_part[b][seg][d].
// ---------------------------------------------------------------------------
__global__ void pool_partial_kernel(const float* __restrict__ X,
                                    float* __restrict__ part) {
    const int b   = blockIdx.z;
    const int seg = blockIdx.y;
    const int d0  = blockIdx.x * 1024 + threadIdx.x * 4;

    const float* p = X + ((size_t)b * LL + (size_t)seg * ROWS_PER_SEG) * DD + d0;
    float a0 = 0.f, a1 = 0.f, a2 = 0.f, a3 = 0.f;
    #pragma unroll 4
    for (int r = 0; r < ROWS_PER_SEG; ++r) {
        const float4 v = *(const float4*)(p + (size_t)r * DD);
        a0 += v.x; a1 += v.y; a2 += v.z; a3 += v.w;
    }
    float* q = part + ((size_t)(b * NSEG + seg)) * DD + d0;
    q[0] = a0; q[1] = a1; q[2] = a2; q[3] = a3;
}

// ---------------------------------------------------------------------------
// Kernel 3: phase-2 pooling -> mean, padded to [16, DD] (rows 8..15 = 0).
// grid = 16*DD/256 blocks.
// ---------------------------------------------------------------------------
__global__ void pool_finalize_kernel(const float* __restrict__ part,
                                     const float* __restrict__ counts,
                                     float* __restrict__ meanP /* [16, DD] */) {
    const int idx = blockIdx.x * 256 + threadIdx.x;  // 0 .. 16*DD-1
    const int m = idx / DD;
    const int d = idx % DD;
    float v = 0.f;
    if (m < BB) {
        float s = 0.f;
        for (int seg = 0; seg < NSEG; ++seg)
            s += part[((size_t)(m * NSEG + seg)) * DD + d];
        const float cnt = counts[m];
        v = s / fmaxf(cnt, 1.0f);
        v = (cnt > 0.f) ? v : 0.f;
    }
    meanP[(size_t)m * DD + d] = v;
}

// ---------------------------------------------------------------------------
// Kernel 4: 16xN = 16xK @ KxN GEMM via V_WMMA_F32_16X16X4_F32.
// One wave32 per 16-wide N-tile; K consumed 4 at a time.
// A: [16,K] row-major, Bm: [K,N] row-major, C: [16,N] row-major.
// ---------------------------------------------------------------------------
__global__ void gemm16_wmma_kernel(const float* __restrict__ A,
                                   const float* __restrict__ Bm,
                                   const float* __restrict__ bias,
                                   float* __restrict__ C,
                                   int K, int N, int applyGelu) {
    const int lane = threadIdx.x & 31;
    const int wave = threadIdx.x >> 5;
    const int tile = blockIdx.x * 4 + wave;
    const int n0 = tile * 16;
    if (n0 >= N) return;                    // wave-uniform: EXEC stays all-1s

    const int ln = lane & 15;               // M for A-frag, N for B-frag
    const int kh = (lane >> 4) << 1;        // 0 (lanes 0-15) or 2 (lanes 16-31)

    v8f c = {};
    for (int k0 = 0; k0 < K; k0 += 4) {
        v2f a, b;
        a[0] = A[(size_t)ln * K + (k0 + kh)];
        a[1] = A[(size_t)ln * K + (k0 + kh + 1)];
        b[0] = Bm[(size_t)(k0 + kh)     * N + n0 + ln];
        b[1] = Bm[(size_t)(k0 + kh + 1) * N + n0 + ln];
        c = __builtin_amdgcn_wmma_f32_16x16x4_f32(
                /*neg_a=*/false, a, /*neg_b=*/false, b,
                /*c_mod=*/(short)0, c, /*reuse_a=*/false, /*reuse_b=*/false);
    }

    const float bv = bias[n0 + ln];
    const int mbase = (lane >> 4) << 3;     // lanes 16-31 hold M = vgpr + 8
    #pragma unroll
    for (int i = 0; i < 8; ++i) {
        float v = c[i] + bv;
        if (applyGelu)
            v = 0.5f * v * (1.0f + erff(v * 0.70710678118654752440f));
        C[(size_t)(mbase + i) * N + n0 + ln] = v;
    }
}

// ---------------------------------------------------------------------------
// Kernel 5: per-row LayerNorm over D2 columns.  grid = 16 rows, 256 thr.
// ---------------------------------------------------------------------------
__global__ void layernorm_kernel(const float* __restrict__ H,
                                 const float* __restrict__ gamma,
                                 const float* __restrict__ beta,
                                 float* __restrict__ HN) {
    __shared__ float red[256];
    __shared__ float s_mu, s_rstd;
    const int m = blockIdx.x;
    const int tid = threadIdx.x;
    const float* row = H + (size_t)m * D2;

    float s = 0.f, ss = 0.f;
    for (int k = tid; k < D2; k += 256) {
        const float v = row[k];
        s += v; ss += v * v;
    }
    red[tid] = s; __syncthreads();
    for (int st = 128; st > 0; st >>= 1) { if (tid < st) red[tid] += red[tid + st]; __syncthreads(); }
    if (tid == 0) s_mu = red[0] / (float)D2;
    __syncthreads();
    red[tid] = ss; __syncthreads();
    for (int st = 128; st > 0; st >>= 1) { if (tid < st) red[tid] += red[tid + st]; __syncthreads(); }
    if (tid == 0) {
        const float mu = s_mu;
        const float var = red[0] / (float)D2 - mu * mu;
        s_rstd = rsqrtf(var + LN_EPS);
    }
    __syncthreads();

    const float mu = s_mu, rstd = s_rstd;
    float* orow = HN + (size_t)m * D2;
    for (int k = tid; k < D2; k += 256)
        orow[k] = (row[k] - mu) * rstd * gamma[k] + beta[k];
}

// ---------------------------------------------------------------------------
// Kernel 6: scatter out[b,:] into first tgt_count[b] target slots, else 0.
// Output [BB, TT, DD]; float4 writes.  grid = BB*TT*DD/4/256 blocks.
// ---------------------------------------------------------------------------
__global__ void scatter_kernel(const float* __restrict__ outm /* [16, DD] */,
                               const float* __restrict__ counts /* tgt at +BB */,
                               float* __restrict__ Y) {
    const int gid = blockIdx.x * 256 + threadIdx.x;       // one float4 each
    const int d4 = gid % (DD / 4);
    const int rest = gid / (DD / 4);
    const int t = rest % TT;
    const int b = rest / TT;
    const bool valid = (float)t < counts[BB + b];

    float4 v = make_float4(0.f, 0.f, 0.f, 0.f);
    if (valid) v = *(const float4*)(outm + (size_t)b * DD + d4 * 4);
    *(float4*)(Y + ((size_t)(b * TT + t)) * DD + d4 * 4) = v;
}

// ---------------------------------------------------------------------------
extern "C" void kernel_launch(void* const* d_in, const int* in_sizes, int n_in,
                              void* d_out, int out_size, void* d_ws, size_t ws_size,
                              hipStream_t stream) {
    const float* ctx   = (const float*)d_in[0];  // [B,L,D]
    const float* tmask = (const float*)d_in[1];  // [B,L]
    const float* cmask = (const float*)d_in[2];  // [B,L]
    const float* W1    = (const float*)d_in[3];  // [D,2D]
    const float* b1    = (const float*)d_in[4];  // [2D]
    const float* gamma = (const float*)d_in[5];  // [2D]
    const float* beta  = (const float*)d_in[6];  // [2D]
    const float* W2    = (const float*)d_in[7];  // [2D,D]
    const float* b2    = (const float*)d_in[8];  // [D]
    float* Y = (float*)d_out;                    // [B,T,D]

    // Workspace layout (floats)
    float* ws      = (float*)d_ws;
    float* w_part  = ws;                                   // BB*NSEG*DD = 524288
    float* w_mean  = w_part + (size_t)BB * NSEG * DD;      // 16*DD      = 32768
    float* w_h     = w_mean + (size_t)16 * DD;             // 16*D2      = 65536
    float* w_hn    = w_h    + (size_t)16 * D2;             // 16*D2      = 65536
    float* w_out   = w_hn   + (size_t)16 * D2;             // 16*DD      = 32768
    float* w_cnt   = w_out  + (size_t)16 * DD;             // 2*BB

    // 1. mask counts
    count_masks_kernel<<<BB, 256, 0, stream>>>(tmask, cmask, w_cnt);

    // 2. pooling phase 1: 512 blocks of coalesced float4 accumulation
    pool_partial_kernel<<<dim3(DD / 1024, NSEG, BB), 256, 0, stream>>>(ctx, w_part);

    // 3. pooling phase 2 -> padded mean [16, DD]
    pool_finalize_kernel<<<(16 * DD) / 256, 256, 0, stream>>>(w_part, w_cnt, w_mean);

    // 4. GEMM1 + bias + exact GELU: [16,DD] @ [DD,D2] -> h [16,D2]
    //    D2/16 = 256 tiles, 4 waves/block -> 64 blocks
    gemm16_wmma_kernel<<<(D2 / 16) / 4, 128, 0, stream>>>(w_mean, W1, b1, w_h, DD, D2, 1);

    // 5. LayerNorm rows
    layernorm_kernel<<<16, 256, 0, stream>>>(w_h, gamma, beta, w_hn);

    // 6. GEMM2 + bias: [16,D2] @ [D2,DD] -> out [16,DD]
    gemm16_wmma_kernel<<<(DD / 16) / 4, 128, 0, stream>>>(w_hn, W2, b2, w_out, D2, DD, 0);

    // 7. scatter into [B,T,D]
    scatter_kernel<<<(BB * TT * (DD / 4)) / 256, 256, 0, stream>>>(w_out, w_cnt, Y);
}